// VoxelAggregation_36112085025262
// MI455X (gfx1250) — compile-verified
//
#include <hip/hip_runtime.h>

// Problem constants (from the reference)
#define GB 4        // batches
#define GD 128      // channels
#define GN 100000   // points per batch (divisible by 4)
#define GG 16       // grid size
#define GV 4096     // voxels = 16^3
#define CHUNK4 1024 // points per scatter block (4 per thread, 256 threads)

typedef int v4i __attribute__((ext_vector_type(4)));

// gfx1250 async global->LDS staging, guarded so compile never breaks.
#if defined(__HIP_DEVICE_COMPILE__) && defined(__gfx1250__)
#  if __has_builtin(__builtin_amdgcn_global_load_async_to_lds_b128)
#    define HAVE_ASYNC_LDS 1
#    pragma message("CDNA5 probe: __builtin_amdgcn_global_load_async_to_lds_b128 AVAILABLE")
#  else
#    pragma message("CDNA5 probe: __builtin_amdgcn_global_load_async_to_lds_b128 NOT available")
#  endif
#endif

// Force native HW FP atomics (global_atomic_add_f32) by asserting the
// accumulator is coarse-grained device memory. Unknown-attribute degrades
// safely to nothing.
#if defined(__has_cpp_attribute)
#  if __has_cpp_attribute(clang::atomic)
#    define ATOMIC_COARSE [[clang::atomic(no_remote_memory, no_fine_grained_memory, ignore_denormal_mode)]]
#  endif
#endif
#ifndef ATOMIC_COARSE
#  define ATOMIC_COARSE
#endif

// ---------------------------------------------------------------------------
// PROBE (never launched; placed FIRST so the disasm snippet shows the exact
// lowering of a coarse-memory f32 atomicAdd: native global_atomic_add_f32 vs
// CAS loop).
// ---------------------------------------------------------------------------
__global__ void probe_fadd_f32(float* __restrict__ p, const float* __restrict__ q) {
    ATOMIC_COARSE {
        atomicAdd(p + (threadIdx.x & 31), q[threadIdx.x]);
    }
}

// ---------------------------------------------------------------------------
// HOT KERNEL. Block = 256 threads, owns (batch b, 1024-point chunk); each
// thread owns a quad of 4 consecutive points. Quad indices staged via b128
// async DMA to LDS, then the 128-channel loop does one b128 row load
// (512 B/wave/issue, fully coalesced) + up to 4 f32 atomics into the
// L2-resident [B,D,V] accumulator (= d_out).
// ---------------------------------------------------------------------------
__global__ void vox_scatter_kernel(const float* __restrict__ feats,
                                   const int*   __restrict__ vidx,
                                   float*       __restrict__ sums) {
    __shared__ int s_vidx[CHUNK4];
    const int nchunks = (GN + CHUNK4 - 1) / CHUNK4;   // 98
    const int blk   = blockIdx.x;
    const int b     = blk / nchunks;
    const int chunk = blk % nchunks;
    const int t     = threadIdx.x;                    // 0..255
    const int p     = chunk * CHUNK4 + 4 * t;         // quad base point
    const bool valid = (p + 3 < GN);                  // quads never straddle N (N%4==0)

    if (valid) {
#ifdef HAVE_ASYNC_LDS
        // gfx1250 async DMA: 4 indices (16 B) straight into LDS (ASYNCcnt).
        __builtin_amdgcn_global_load_async_to_lds_b128(
            (__attribute__((address_space(1))) v4i*)(vidx + (size_t)b * GN + p),
            (__attribute__((address_space(3))) v4i*)(&s_vidx[4 * t]),
            0, 0);
#else
        ((int4*)s_vidx)[t] = *(const int4*)(vidx + (size_t)b * GN + p);
#endif
    } else {
        ((int4*)s_vidx)[t] = make_int4(-1, -1, -1, -1);
    }
#ifdef HAVE_ASYNC_LDS
#  if __has_builtin(__builtin_amdgcn_s_wait_asynccnt)
    __builtin_amdgcn_s_wait_asynccnt(0);
#  else
    asm volatile("s_wait_asynccnt 0" ::: "memory");
#  endif
#endif
    __syncthreads();

    const int4 vv = ((const int4*)s_vidx)[t];
    if ((vv.x < 0) && (vv.y < 0) && (vv.z < 0) && (vv.w < 0)) return;

    const float* fq = feats + (size_t)b * GD * GN + p;   // quad column, stride N over d
    float*       sb = sums  + (size_t)b * GD * GV;       // batch accumulator base

    ATOMIC_COARSE {
#pragma unroll 2
        for (int d = 0; d < GD; ++d) {
            if ((d & 7) == 0) {
                // prefetch 8 rows ahead (speculative; OOB prefetch dropped by HW)
                __builtin_prefetch(fq + (size_t)(d + 8) * GN, 0, 0);
            }
            const float4 f = *(const float4*)(fq + (size_t)d * GN);
            const size_t off = (size_t)d * GV;
            if (vv.x >= 0) atomicAdd(sb + off + vv.x, f.x);
            if (vv.y >= 0) atomicAdd(sb + off + vv.y, f.y);
            if (vv.z >= 0) atomicAdd(sb + off + vv.z, f.z);
            if (vv.w >= 0) atomicAdd(sb + off + vv.w, f.w);
        }
    }
}

// ---------------------------------------------------------------------------
// Kernel: zero the accumulator (d_out reused as [B,D,V] sums) and counts.
// ---------------------------------------------------------------------------
__global__ void vox_zero_kernel(float* __restrict__ out, int* __restrict__ counts) {
    size_t i      = (size_t)blockIdx.x * blockDim.x + threadIdx.x;
    size_t stride = (size_t)gridDim.x * blockDim.x;
    const size_t nout = (size_t)GB * GD * GV;
    for (size_t k = i; k < nout; k += stride) out[k] = 0.0f;
    const size_t ncnt = (size_t)GB * GV;
    for (size_t k = i; k < ncnt; k += stride) counts[k] = 0;
}

// ---------------------------------------------------------------------------
// Kernel: per-point voxel index (+mask fold) and voxel counts.
// vidx = -1 for masked-out points: they contribute 0 to sums AND counts.
// ---------------------------------------------------------------------------
__global__ void vox_index_kernel(const float* __restrict__ xyz,
                                 const int*   __restrict__ mask,
                                 int*         __restrict__ vidx,
                                 int*         __restrict__ counts) {
    int i = blockIdx.x * blockDim.x + threadIdx.x;   // over B*N
    if (i >= GB * GN) return;
    const size_t base = (size_t)i * 3;
    float x = xyz[base + 0];
    float y = xyz[base + 1];
    float z = xyz[base + 2];
    int vx = min(max((int)(x * (float)GG), 0), GG - 1);
    int vy = min(max((int)(y * (float)GG), 0), GG - 1);
    int vz = min(max((int)(z * (float)GG), 0), GG - 1);
    int flat = vz * (GG * GG) + vy * GG + vx;
    int b = i / GN;
    if (mask[i] != 0) {
        vidx[i] = flat;
        ATOMIC_COARSE {
            atomicAdd(&counts[b * GV + flat], 1);
        }
    } else {
        vidx[i] = -1;
    }
}

// ---------------------------------------------------------------------------
// Kernel: in-place mean: out[b,d,v] /= max(count[b,v],1).
// Fully coalesced load+store; counts lookups hit L1/L2.
// ---------------------------------------------------------------------------
__global__ void vox_finalize_kernel(float* __restrict__ out,
                                    const int* __restrict__ counts) {
    size_t i      = (size_t)blockIdx.x * blockDim.x + threadIdx.x;
    size_t stride = (size_t)gridDim.x * blockDim.x;
    const size_t ntot = (size_t)GB * GD * GV;
    for (size_t k = i; k < ntot; k += stride) {
        int v = (int)(k % GV);
        int b = (int)(k / ((size_t)GD * GV));
        int c = counts[b * GV + v];
        if (c < 1) c = 1;
        out[k] = out[k] / (float)c;
    }
}

// ---------------------------------------------------------------------------
extern "C" void kernel_launch(void* const* d_in, const int* in_sizes, int n_in,
                              void* d_out, int out_size, void* d_ws, size_t ws_size,
                              hipStream_t stream) {
    const float* feats = (const float*)d_in[0];   // [B, D, N] f32
    const float* xyz   = (const float*)d_in[1];   // [B, N, 3] f32
    const int*   mask  = (const int*)  d_in[2];   // [B, N] i32
    float*       out   = (float*)d_out;           // [B, D, V] f32 (also the accumulator)

    // Workspace layout: counts [B*V] int, vidx [B*N] int  (~1.7 MB)
    int* counts = (int*)d_ws;
    int* vidx   = counts + (size_t)GB * GV;

    // 1) zero accumulator + counts
    vox_zero_kernel<<<2048, 256, 0, stream>>>(out, counts);

    // 2) voxel indices + counts
    {
        int total  = GB * GN;
        int blocks = (total + 255) / 256;
        vox_index_kernel<<<blocks, 256, 0, stream>>>(xyz, mask, vidx, counts);
    }

    // 3) scatter-add features (bandwidth-bound streamer)
    {
        const int nchunks = (GN + CHUNK4 - 1) / CHUNK4;   // 98
        vox_scatter_kernel<<<GB * nchunks, 256, 0, stream>>>(feats, vidx, out);
    }

    // 4) divide by counts in place
    vox_finalize_kernel<<<2048, 256, 0, stream>>>(out, counts);
}